// PointCloudRefinement_40802189312464
// MI455X (gfx1250) — compile-verified
//
#include <hip/hip_runtime.h>
#include <hip/hip_bf16.h>
#include <math.h>

typedef __attribute__((ext_vector_type(2))) float v2f;
typedef __attribute__((ext_vector_type(8))) float v8f;

#define WAVESZ 32
#define WAVES_PER_BLOCK 8
#define ROWS_PER_WAVE 16
// exp(10*x) == exp2(K*x),  K = 10 * log2(e)  (TEMPERATURE = 0.1)
#define K_EXP2 14.4269504088896340736f

// ---------------------------------------------------------------------------
// Fused tiled pairwise-distance row-reduction.
//   P: (B, M, 3) rows owned by this kernel, G: (B, N, 3) columns streamed.
//   Each wave owns 16 rows, loops over N in 16-column tiles.
//   V_WMMA_F32_16X16X4_F32 computes all 256 dot products of a tile.
//   WITH_EMD: d = v_sqrt(d2); streaming branchless softmin (online-softmax)
//             with raw v_exp2 weights; running min doubles as Chamfer term.
//   !WITH_EMD: accumulate min(d2) only; sqrt once per row at the end
//             (min(sqrt(x)) == sqrt(min(x))) -> no TRANS ops in the loop.
//   No atomics, no materialized distance matrix; inputs stay L2-resident.
// ---------------------------------------------------------------------------
template <bool WITH_EMD>
__global__ __launch_bounds__(256) void pcr_tile_rowreduce(
    const float* __restrict__ P, const float* __restrict__ G, int M, int N,
    float* __restrict__ rowmin_out,  // (B, M)
    float* __restrict__ emd_out)     // (B, M) (unused if !WITH_EMD)
{
  const int lane = threadIdx.x & (WAVESZ - 1);
  const int wv   = threadIdx.x >> 5;
  const int b    = blockIdx.y;
  const int m0   = (blockIdx.x * WAVES_PER_BLOCK + wv) * ROWS_PER_WAVE;
  const int hi   = lane >> 4;   // 0: lanes 0-15, 1: lanes 16-31
  const int l16  = lane & 15;

  const float* Pb = P + (size_t)b * M * 3;
  const float* Gb = G + (size_t)b * N * 3;

  // --- A fragment: 16x4 f32, rows m0..m0+15, K = (x, y, z, 0) -------------
  // ISA layout: lanes 0-15 hold K=0 (VGPR0) / K=1 (VGPR1);
  //             lanes 16-31 hold K=2 / K=3.  Row index = lane & 15.
  const float* pr = Pb + (size_t)(m0 + l16) * 3;
  const float rx = pr[0], ry = pr[1], rz = pr[2];
  v2f afrag;
  afrag.x = hi ? rz : rx;
  afrag.y = hi ? 0.0f : ry;
  const float p2row = rx * rx + ry * ry + rz * rz;

  // Slot v of the C fragment is row m0 + v + 8*hi -> fetch its |p|^2 via
  // width-16 shuffle (wave32-native cross-lane move).
  float p2v[8];
#pragma unroll
  for (int v = 0; v < 8; ++v) p2v[v] = __shfl(p2row, v + 8 * hi, 16);

  float mn[8], S[8], W[8];
#pragma unroll
  for (int v = 0; v < 8; ++v) {
    mn[v] = __builtin_inff();
    S[v] = 0.0f;
    W[v] = 0.0f;
  }

  for (int n0 = 0; n0 < N; n0 += 16) {
    // g tile: column n0 + l16 (duplicated across the two half-waves)
    const float* gp = Gb + (size_t)(n0 + l16) * 3;
    const float gx = gp[0], gy = gp[1], gz = gp[2];
    if (n0 + 16 < N)  // CDNA5 global_prefetch_b8 of next tile
      __builtin_prefetch(Gb + (size_t)(n0 + 16 + l16) * 3, 0, 0);
    const float g2 = gx * gx + gy * gy + gz * gz;

    v2f bfrag;  // B 4x16 f32: K=0/K=1 in lanes 0-15, K=2/K=3 in lanes 16-31
    bfrag.x = hi ? gz : gx;
    bfrag.y = hi ? 0.0f : gy;

    v8f c = {};
    c = __builtin_amdgcn_wmma_f32_16x16x4_f32(
        /*neg_a=*/false, afrag, /*neg_b=*/false, bfrag,
        /*c_mod=*/(short)0, c, /*reuse_a=*/false, /*reuse_b=*/false);

#pragma unroll
    for (int v = 0; v < 8; ++v) {
      const float d2 = fmaxf(p2v[v] + g2 - 2.0f * c[v], 0.0f);
      if (WITH_EMD) {
        const float d = __builtin_amdgcn_sqrtf(d2);  // raw v_sqrt_f32
        // branchless streaming softmin:  S = sum exp(10*(mn - d_i)),
        // W = sum d_i * exp(10*(mn - d_i)), rescaled when min improves.
        const float nm = fminf(mn[v], d);
        const float r  = __builtin_amdgcn_exp2f(K_EXP2 * (nm - mn[v]));
        const float e  = __builtin_amdgcn_exp2f(K_EXP2 * (nm - d));
        S[v]  = fmaf(S[v], r, e);
        W[v]  = fmaf(W[v], r, d * e);
        mn[v] = nm;
      } else {
        mn[v] = fminf(mn[v], d2);  // min of squared dists; sqrt at the end
      }
    }
  }

  // --- reduce each slot across the 16 lanes of its half-wave --------------
#pragma unroll
  for (int v = 0; v < 8; ++v) {
#pragma unroll
    for (int off = 1; off < 16; off <<= 1) {
      const float mn2 = __shfl_xor(mn[v], off, 16);
      if (WITH_EMD) {
        const float S2 = __shfl_xor(S[v], off, 16);
        const float W2 = __shfl_xor(W[v], off, 16);
        const float m3 = fminf(mn[v], mn2);
        const float r1 = __builtin_amdgcn_exp2f(K_EXP2 * (m3 - mn[v]));
        const float r2 = __builtin_amdgcn_exp2f(K_EXP2 * (m3 - mn2));
        S[v]  = fmaf(S[v], r1, S2 * r2);
        W[v]  = fmaf(W[v], r1, W2 * r2);
        mn[v] = m3;
      } else {
        mn[v] = fminf(mn[v], mn2);
      }
    }
  }

  // lanes with l16 < 8 write row m0 + l16 + 8*hi (select slot branchlessly)
  if (l16 < 8) {
    float omn = mn[0], oS = S[0], oW = W[0];
#pragma unroll
    for (int v = 1; v < 8; ++v)
      if (l16 == v) { omn = mn[v]; oS = S[v]; oW = W[v]; }
    const int m = m0 + l16 + 8 * hi;
    if (WITH_EMD) {
      rowmin_out[(size_t)b * M + m] = omn;           // already a distance
      emd_out[(size_t)b * M + m]    = oW / oS;
    } else {
      rowmin_out[(size_t)b * M + m] = __builtin_amdgcn_sqrtf(omn);
    }
  }
}

// ---------------------------------------------------------------------------
// Final reduction: means of row results + ddof=1 variance of corrections.
// Single block; work is O(B*(M+N) + B*M*3) ~ 130K elements -> negligible.
// ---------------------------------------------------------------------------
__global__ __launch_bounds__(256) void pcr_finalize(
    const float* __restrict__ rmin_p, const float* __restrict__ emd_row,
    const float* __restrict__ rmin_g, const float* __restrict__ pred,
    const float* __restrict__ pseudo, int B, int M, int N,
    float* __restrict__ out)
{
  __shared__ float red[256];
  const int tid = threadIdx.x;

  float s_cd1 = 0.0f, s_emd = 0.0f, s_cd2 = 0.0f;
  for (int i = tid; i < B * M; i += 256) {
    s_cd1 += rmin_p[i];
    s_emd += emd_row[i];
  }
  for (int i = tid; i < B * N; i += 256) s_cd2 += rmin_g[i];

  float sc[6], sq[6];
#pragma unroll
  for (int k = 0; k < 6; ++k) { sc[k] = 0.0f; sq[k] = 0.0f; }
  const int total = B * M * 3;
  for (int i = tid; i < total; i += 256) {
    const float c = pred[i] - pseudo[i];
    const int bucket = (i / (M * 3)) * 3 + (i % 3);
#pragma unroll
    for (int k = 0; k < 6; ++k)
      if (bucket == k) { sc[k] += c; sq[k] += c * c; }
  }

  auto block_sum = [&](float v) -> float {
    red[tid] = v;
    __syncthreads();
    for (int s = 128; s > 0; s >>= 1) {
      if (tid < s) red[tid] += red[tid + s];
      __syncthreads();
    }
    const float r = red[0];
    __syncthreads();
    return r;
  };

  const float cd1 = block_sum(s_cd1);
  const float emd = block_sum(s_emd);
  const float cd2 = block_sum(s_cd2);
  float vsum = 0.0f;
#pragma unroll
  for (int k = 0; k < 6; ++k) {
    const float csum = block_sum(sc[k]);
    const float csq  = block_sum(sq[k]);
    if (tid == 0)
      vsum += (csq - csum * csum / (float)M) / (float)(M - 1);
  }

  if (tid == 0) {
    const float cd     = cd1 / (float)(B * M) + cd2 / (float)(B * N);
    const float emd_m  = emd / (float)(B * M);
    const float smooth = vsum / 6.0f;
    out[0] = cd + emd_m + 0.1f * smooth;  // W_CHAMFER=1, W_EMD=1, W_SMOOTH=0.1
  }
}

// ---------------------------------------------------------------------------
extern "C" void kernel_launch(void* const* d_in, const int* in_sizes, int n_in,
                              void* d_out, int out_size, void* d_ws,
                              size_t ws_size, hipStream_t stream)
{
  (void)n_in; (void)out_size; (void)ws_size;
  const float* pred   = (const float*)d_in[0];  // (B, M, 3)
  const float* gt     = (const float*)d_in[1];  // (B, N, 3)
  const float* pseudo = (const float*)d_in[2];  // (B, M, 3)

  const int B = 2;
  const int M = in_sizes[0] / (B * 3);  // 8192
  const int N = in_sizes[1] / (B * 3);  // 8192

  float* rmin_p = (float*)d_ws;            // B*M
  float* emd_r  = rmin_p + (size_t)B * M;  // B*M
  float* rmin_g = emd_r + (size_t)B * M;   // B*N

  const dim3 blk(256);
  const dim3 grdA(M / (WAVES_PER_BLOCK * ROWS_PER_WAVE), B);
  const dim3 grdB(N / (WAVES_PER_BLOCK * ROWS_PER_WAVE), B);

  // Pass 1: rows = pred, cols = gt -> row minima (Chamfer term 1) + EMD
  pcr_tile_rowreduce<true><<<grdA, blk, 0, stream>>>(pred, gt, M, N, rmin_p, emd_r);
  // Pass 2: rows = gt, cols = pred -> row minima of transpose (Chamfer term 2)
  pcr_tile_rowreduce<false><<<grdB, blk, 0, stream>>>(gt, pred, N, M, rmin_g, nullptr);
  // Combine
  pcr_finalize<<<dim3(1), blk, 0, stream>>>(rmin_p, emd_r, rmin_g, pred, pseudo,
                                            B, M, N, (float*)d_out);
}